// SAGEScheduler_75582834475359
// MI455X (gfx1250) — compile-verified
//
#include <hip/hip_runtime.h>
#include <hip/hip_bf16.h>

// ---------------- problem constants ----------------
#define N_NODES 5000
#define NPAD    5120      // multiple of 256: WMMA row tiles never straddle
#define N_EDGES 160000
#define HID     64
#define NODE_F  4
#define TASK_F  4
#define BATCH   512

typedef __attribute__((ext_vector_type(2))) float v2f;
typedef __attribute__((ext_vector_type(8))) float v8f;

// ---------------- workspace layout (floats) ----------------
// [deg 5120][agg1 5120*4][agg2 5120*64]  <- zeroed region (contiguous)
// [rdeg 5120][h1][h2][hpart][tpart 512*64]
#define OFF_DEG    0
#define OFF_AGG1   (OFF_DEG  + NPAD)
#define OFF_AGG2   (OFF_AGG1 + NPAD * NODE_F)
#define ZERO_COUNT (OFF_AGG2 + NPAD * HID)          // 353,280 floats
#define OFF_RDEG   ZERO_COUNT
#define OFF_H1     (OFF_RDEG  + NPAD)
#define OFF_H2     (OFF_H1    + NPAD * HID)
#define OFF_HPART  (OFF_H2    + NPAD * HID)
#define OFF_TPART  (OFF_HPART + NPAD * HID)

// ---------------- kernels ----------------

__global__ void k_zero(float* __restrict__ p, int n) {
    int i = blockIdx.x * blockDim.x + threadIdx.x;
    if (i < n) p[i] = 0.0f;
}

// scatter layer 1: deg[dst] += 1 ; agg1[dst][0..3] += x[src][0..3]
__global__ void k_scatter1(const int* __restrict__ ei, const float* __restrict__ x,
                           float* __restrict__ agg1, float* __restrict__ deg) {
    int e = blockIdx.x * blockDim.x + threadIdx.x;
    if (e >= N_EDGES) return;
    int s = ei[e];              // row 0 = src
    int d = ei[N_EDGES + e];    // row 1 = dst
    atomicAdd(&deg[d], 1.0f);
#pragma unroll
    for (int k = 0; k < NODE_F; ++k)
        atomicAdd(&agg1[d * NODE_F + k], x[s * NODE_F + k]);
}

__global__ void k_rdeg(const float* __restrict__ deg, float* __restrict__ rdeg) {
    int i = blockIdx.x * blockDim.x + threadIdx.x;
    if (i < NPAD) rdeg[i] = 1.0f / fmaxf(deg[i], 1.0f);
}

// SAGE layer 1: h1 = relu(mean1 @ Wl1.T + x @ Wr1.T + bl1)   (K = NODE_F = 4: one WMMA per chain)
// One wave (32 threads) per 16x16 output tile.
__global__ void k_sage1(const float* __restrict__ agg1, const float* __restrict__ rdeg,
                        const float* __restrict__ x,
                        const float* __restrict__ Wl1, const float* __restrict__ bl1,
                        const float* __restrict__ Wr1, float* __restrict__ h1) {
    const int row0 = blockIdx.x * 16;
    const int col0 = blockIdx.y * 16;
    const int lane = threadIdx.x;       // 0..31, full wave, EXEC all ones
    const int m    = lane & 15;
    const int half = lane >> 4;
    const int kb   = half * 2;          // A/B fragment K-pair base
    const int row  = row0 + m;
    const float rd = rdeg[row];

    v2f am, ax, bl, br;
    am.x = agg1[row * NODE_F + kb]     * rd;
    am.y = agg1[row * NODE_F + kb + 1] * rd;
    // x only has N_NODES real rows; padded rows contribute deterministic zeros
    ax.x = (row < N_NODES) ? x[row * NODE_F + kb]     : 0.0f;
    ax.y = (row < N_NODES) ? x[row * NODE_F + kb + 1] : 0.0f;
    bl.x = Wl1[(col0 + m) * NODE_F + kb];
    bl.y = Wl1[(col0 + m) * NODE_F + kb + 1];
    br.x = Wr1[(col0 + m) * NODE_F + kb];
    br.y = Wr1[(col0 + m) * NODE_F + kb + 1];

    v8f acc = {};
    acc = __builtin_amdgcn_wmma_f32_16x16x4_f32(false, am, false, bl, (short)0, acc, false, false);
    acc = __builtin_amdgcn_wmma_f32_16x16x4_f32(false, ax, false, br, (short)0, acc, false, false);

#pragma unroll
    for (int i = 0; i < 8; ++i) {
        int r = row0 + i + half * 8;
        int c = col0 + m;
        h1[r * HID + c] = fmaxf(acc[i] + bl1[c], 0.0f);
    }
}

// scatter layer 2: agg2[dst][f] += h1[src][f], one thread per (edge, feature)
__global__ void k_scatter2(const int* __restrict__ ei, const float* __restrict__ h1,
                           float* __restrict__ agg2) {
    long long idx = (long long)blockIdx.x * blockDim.x + threadIdx.x;
    if (idx >= (long long)N_EDGES * HID) return;
    int e = (int)(idx >> 6);
    int f = (int)(idx & 63);
    int s = ei[e];
    int d = ei[N_EDGES + e];
    atomicAdd(&agg2[d * HID + f], h1[s * HID + f]);
}

// SAGE layer 2: h2 = relu(mean2 @ Wl2.T + h1 @ Wr2.T + bl2)   K=64 -> 16 k-slices x 2 chains
__global__ void k_sage2(const float* __restrict__ agg2, const float* __restrict__ rdeg,
                        const float* __restrict__ h1,
                        const float* __restrict__ Wl2, const float* __restrict__ bl2,
                        const float* __restrict__ Wr2, float* __restrict__ h2) {
    const int row0 = blockIdx.x * 16;
    const int col0 = blockIdx.y * 16;
    const int lane = threadIdx.x;
    const int m    = lane & 15;
    const int half = lane >> 4;
    const int kb   = half * 2;
    const int row  = row0 + m;
    const float rd = rdeg[row];

    v8f acc = {};
#pragma unroll 4
    for (int k0 = 0; k0 < HID; k0 += 4) {
        v2f am, ah, bl, br;
        am.x = agg2[row * HID + k0 + kb]     * rd;
        am.y = agg2[row * HID + k0 + kb + 1] * rd;
        ah.x = h1[row * HID + k0 + kb];
        ah.y = h1[row * HID + k0 + kb + 1];
        bl.x = Wl2[(col0 + m) * HID + k0 + kb];
        bl.y = Wl2[(col0 + m) * HID + k0 + kb + 1];
        br.x = Wr2[(col0 + m) * HID + k0 + kb];
        br.y = Wr2[(col0 + m) * HID + k0 + kb + 1];
        acc = __builtin_amdgcn_wmma_f32_16x16x4_f32(false, am, false, bl, (short)0, acc, false, false);
        acc = __builtin_amdgcn_wmma_f32_16x16x4_f32(false, ah, false, br, (short)0, acc, false, false);
    }
#pragma unroll
    for (int i = 0; i < 8; ++i) {
        int r = row0 + i + half * 8;
        int c = col0 + m;
        h2[r * HID + c] = fmaxf(acc[i] + bl2[c], 0.0f);
    }
}

// h_part = h2 @ Wc1[:, :64].T      (Wc1 is [64,128] row-major; no bias, no relu)
__global__ void k_hpart(const float* __restrict__ h2, const float* __restrict__ Wc1,
                        float* __restrict__ hpart) {
    const int row0 = blockIdx.x * 16;
    const int col0 = blockIdx.y * 16;
    const int lane = threadIdx.x;
    const int m    = lane & 15;
    const int half = lane >> 4;
    const int kb   = half * 2;
    const int row  = row0 + m;

    v8f acc = {};
#pragma unroll 4
    for (int k0 = 0; k0 < HID; k0 += 4) {
        v2f a, b;
        a.x = h2[row * HID + k0 + kb];
        a.y = h2[row * HID + k0 + kb + 1];
        b.x = Wc1[(col0 + m) * (2 * HID) + k0 + kb];
        b.y = Wc1[(col0 + m) * (2 * HID) + k0 + kb + 1];
        acc = __builtin_amdgcn_wmma_f32_16x16x4_f32(false, a, false, b, (short)0, acc, false, false);
    }
#pragma unroll
    for (int i = 0; i < 8; ++i) {
        int r = row0 + i + half * 8;
        int c = col0 + m;
        hpart[r * HID + c] = acc[i];
    }
}

// task encoder + t-part of classifier: one block (64 threads) per batch row
// t = relu(task @ Wt1.T + bt1) @ Wt2.T + bt2 ;  tpart = t @ Wc1[:,64:].T + bc1
__global__ void k_task(const float* __restrict__ task, const float* __restrict__ Wt1,
                       const float* __restrict__ bt1, const float* __restrict__ Wt2,
                       const float* __restrict__ bt2, const float* __restrict__ Wc1,
                       const float* __restrict__ bc1, float* __restrict__ tpart) {
    __shared__ float u[HID];
    __shared__ float t[HID];
    const int b = blockIdx.x;
    const int h = threadIdx.x;   // 0..63

    float acc = bt1[h];
#pragma unroll
    for (int k = 0; k < TASK_F; ++k) acc += task[b * TASK_F + k] * Wt1[h * TASK_F + k];
    u[h] = fmaxf(acc, 0.0f);
    __syncthreads();

    float acc2 = bt2[h];
#pragma unroll 8
    for (int k = 0; k < HID; ++k) acc2 += u[k] * Wt2[h * HID + k];
    t[h] = acc2;
    __syncthreads();

    float acc3 = bc1[h];
#pragma unroll 8
    for (int k = 0; k < HID; ++k) acc3 += t[k] * Wc1[h * (2 * HID) + HID + k];
    tpart[b * HID + h] = acc3;
}

// fused classifier: scores[b,n] = sum_h relu(hpart[n,h] + tpart[b,h]) * Wc2[h] + bc2
// block = 256 threads (1 node each), one batch row per blockIdx.y; tpart row + Wc2 in LDS
__global__ void k_classify(const float* __restrict__ hpart, const float* __restrict__ tpart,
                           const float* __restrict__ Wc2, const float* __restrict__ bc2,
                           float* __restrict__ out) {
    __shared__ float tp[HID];
    __shared__ float w2[HID];
    const int b = blockIdx.y;
    const int n = blockIdx.x * 256 + threadIdx.x;
    if (threadIdx.x < HID) {
        tp[threadIdx.x] = tpart[b * HID + threadIdx.x];
        w2[threadIdx.x] = Wc2[threadIdx.x];
    }
    __syncthreads();
    if (n >= N_NODES) return;

    const float4* hp = (const float4*)(hpart + (size_t)n * HID);
    float acc = bc2[0];
#pragma unroll
    for (int j = 0; j < HID / 4; ++j) {
        float4 v = hp[j];
        int h = j * 4;
        acc += fmaxf(v.x + tp[h + 0], 0.0f) * w2[h + 0];
        acc += fmaxf(v.y + tp[h + 1], 0.0f) * w2[h + 1];
        acc += fmaxf(v.z + tp[h + 2], 0.0f) * w2[h + 2];
        acc += fmaxf(v.w + tp[h + 3], 0.0f) * w2[h + 3];
    }
    out[(size_t)b * N_NODES + n] = acc;
}

// ---------------- launcher ----------------
extern "C" void kernel_launch(void* const* d_in, const int* in_sizes, int n_in,
                              void* d_out, int out_size, void* d_ws, size_t ws_size,
                              hipStream_t stream) {
    const float* x    = (const float*)d_in[0];
    const int*   ei   = (const int*)d_in[1];   // edge_index (int32 under default JAX x64-off)
    const float* task = (const float*)d_in[2];
    const float* Wl1  = (const float*)d_in[3];
    const float* bl1  = (const float*)d_in[4];
    const float* Wr1  = (const float*)d_in[5];
    const float* Wl2  = (const float*)d_in[6];
    const float* bl2  = (const float*)d_in[7];
    const float* Wr2  = (const float*)d_in[8];
    const float* Wt1  = (const float*)d_in[9];
    const float* bt1  = (const float*)d_in[10];
    const float* Wt2  = (const float*)d_in[11];
    const float* bt2  = (const float*)d_in[12];
    const float* Wc1  = (const float*)d_in[13];
    const float* bc1  = (const float*)d_in[14];
    const float* Wc2  = (const float*)d_in[15];
    const float* bc2  = (const float*)d_in[16];
    float* out = (float*)d_out;

    float* ws    = (float*)d_ws;
    float* deg   = ws + OFF_DEG;
    float* agg1  = ws + OFF_AGG1;
    float* agg2  = ws + OFF_AGG2;
    float* rdeg  = ws + OFF_RDEG;
    float* h1    = ws + OFF_H1;
    float* h2    = ws + OFF_H2;
    float* hpart = ws + OFF_HPART;
    float* tpart = ws + OFF_TPART;

    // 1) zero accumulators (deg, agg1, agg2 are contiguous)
    k_zero<<<(ZERO_COUNT + 255) / 256, 256, 0, stream>>>(ws, ZERO_COUNT);
    // 2) layer-1 scatter
    k_scatter1<<<(N_EDGES + 255) / 256, 256, 0, stream>>>(ei, x, agg1, deg);
    // 3) reciprocal degree
    k_rdeg<<<(NPAD + 255) / 256, 256, 0, stream>>>(deg, rdeg);
    // 4) SAGE layer 1 (WMMA)
    k_sage1<<<dim3(NPAD / 16, HID / 16), 32, 0, stream>>>(agg1, rdeg, x, Wl1, bl1, Wr1, h1);
    // 5) layer-2 scatter
    {
        long long total = (long long)N_EDGES * HID;
        k_scatter2<<<(unsigned)((total + 255) / 256), 256, 0, stream>>>(ei, h1, agg2);
    }
    // 6) SAGE layer 2 (WMMA)
    k_sage2<<<dim3(NPAD / 16, HID / 16), 32, 0, stream>>>(agg2, rdeg, h1, Wl2, bl2, Wr2, h2);
    // 7) h_part (WMMA)
    k_hpart<<<dim3(NPAD / 16, HID / 16), 32, 0, stream>>>(h2, Wc1, hpart);
    // 8) task encoder + t_part
    k_task<<<BATCH, HID, 0, stream>>>(task, Wt1, bt1, Wt2, bt2, Wc1, bc1, tpart);
    // 9) fused classifier
    k_classify<<<dim3((N_NODES + 255) / 256, BATCH), 256, 0, stream>>>(hpart, tpart, Wc2, bc2, out);
}